// PositionEncoder_34050500722884
// MI455X (gfx1250) — compile-verified
//
#include <hip/hip_runtime.h>
#include <math.h>

#define B_   8
#define N_   4096
#define M_   1024
#define K_   20
#define C_   128
#define C8_  16
#define C4_  32
#define EPS_ 1e-5f

typedef __attribute__((ext_vector_type(2))) float v2f;
typedef __attribute__((ext_vector_type(8))) float v8f;

__device__ __forceinline__ float gelu_exact(float x) {
    return 0.5f * x * (1.0f + erff(x * 0.70710678118654752f));
}

// ---------------------------------------------------------------------------
// Kernel 1: ball query. One wave32 per (b, m) row of dist. Ordered compaction
// of "first K indices with dist <= radius" via ballot + prefix-popcount, with
// scalar early exit once K matches are found (dist is ~25% hit rate, so we
// touch only a few 32-element chunks of the 4096-element row).
// ---------------------------------------------------------------------------
__global__ __launch_bounds__(256)
void ball_query_kernel(const float* __restrict__ dist,
                       const float* __restrict__ radius_p,
                       int* __restrict__ idx_out) {
    const float radius = radius_p[0];
    const int lane = threadIdx.x & 31;
    const int wave = threadIdx.x >> 5;
    const int row  = blockIdx.x * 8 + wave;          // 0 .. B*M-1
    const float* __restrict__ drow = dist + (size_t)row * N_;
    int* __restrict__ irow = idx_out + (size_t)row * K_;

    int count = 0;
    int first = N_;   // sentinel: no match -> gather clamps to N-1 (jnp.take clip)
    for (int base = 0; base < N_ && count < K_; base += 32) {
        float d = drow[base + lane];
        bool p = (d <= radius);
        unsigned mask = (unsigned)__ballot(p);        // wave32: low 32 bits
        if (mask != 0u && count == 0) first = base + (__ffs(mask) - 1);
        if (p) {
            int pos = count + __popc(mask & ((1u << lane) - 1u));
            if (pos < K_) irow[pos] = base + lane;
        }
        count += __popc(mask);
    }
    // pad unwritten slots [count, K) with the first found index
    if (lane >= count && lane < K_) irow[lane] = first;
}

// ---------------------------------------------------------------------------
// Kernel 2: gather + feature build (9-d) + conv1 (16x9) + BN + exact GELU +
// max over K. One thread per (b, m). Output x16 laid out (B, 16, M) so the
// WMMA kernel can load contiguous 16-wide B-tiles per channel row.
// ---------------------------------------------------------------------------
__global__ __launch_bounds__(256)
void group_conv1_kernel(const float* __restrict__ xyz,
                        const float* __restrict__ centroid,
                        const int* __restrict__ idx,
                        const float* __restrict__ W1,
                        const float* __restrict__ b1,
                        const float* __restrict__ g1, const float* __restrict__ bb1,
                        const float* __restrict__ mm1, const float* __restrict__ vv1,
                        float* __restrict__ x16) {
    __shared__ float sW1[C8_ * 9];
    __shared__ float sB1[C8_], sScale[C8_], sShift[C8_];
    const int t = threadIdx.x;
    if (t < C8_ * 9) sW1[t] = W1[t];
    if (t < C8_) {
        float sc = g1[t] * rsqrtf(vv1[t] + EPS_);
        sB1[t]    = b1[t];
        sScale[t] = sc;
        sShift[t] = bb1[t] - mm1[t] * sc;
    }
    __syncthreads();

    const int gid = blockIdx.x * blockDim.x + t;     // 0 .. B*M-1
    const int b = gid / M_;
    const int m = gid % M_;
    const float* __restrict__ xb = xyz + (size_t)b * 3 * N_;
    const float cx = centroid[((size_t)b*3 + 0)*M_ + m];
    const float cy = centroid[((size_t)b*3 + 1)*M_ + m];
    const float cz = centroid[((size_t)b*3 + 2)*M_ + m];
    const int* __restrict__ irow = idx + (size_t)gid * K_;

    float mx[C8_];
    #pragma unroll
    for (int o = 0; o < C8_; ++o) mx[o] = -INFINITY;

    for (int k = 0; k < K_; ++k) {
        int n = irow[k];
        n = (n < N_) ? n : (N_ - 1);                  // jnp.take 'clip'
        float gx = xb[0*N_ + n], gy = xb[1*N_ + n], gz = xb[2*N_ + n];
        float f[9] = {cx, cy, cz, gx, gy, gz, gx - cx, gy - cy, gz - cz};
        #pragma unroll
        for (int o = 0; o < C8_; ++o) {
            float s = sB1[o];
            #pragma unroll
            for (int c = 0; c < 9; ++c) s = fmaf(sW1[o*9 + c], f[c], s);
            s = s * sScale[o] + sShift[o];
            s = gelu_exact(s);
            mx[o] = fmaxf(mx[o], s);
        }
    }
    #pragma unroll
    for (int o = 0; o < C8_; ++o)
        x16[((size_t)b * C8_ + o) * M_ + m] = mx[o];
}

// ---------------------------------------------------------------------------
// Kernel 3: fused MLP2 + MLP3 on WMMA f32 16x16x4, one wave32 per 16-wide
// m-tile. GEMM1: h(32x16) = W2(32x16) @ x16tile(16x16) in 2 D-tiles x 4
// k-steps. BN2 + GELU applied in D-layout, tile staged to LDS to convert
// D-layout -> B-layout, then GEMM2: out(128x16) = W3(128x32) @ h(32x16) in
// 8 o-tiles x 8 k-steps. BN3 + store.
//
// Wave32 operand layouts (ISA 7.12.2):
//   A 16x4 f32 : lane L holds row M=L%16; lanes 0-15 -> K={k0,k0+1} in
//                {v.x,v.y}; lanes 16-31 -> K={k0+2,k0+3}.
//   B 4x16 f32 : lane L holds col N=L%16; same K split by lane half.
//   C/D 16x16  : 8 VGPRs; VGPR r = row (r | r+8 by lane half), lane%16 = col.
// ---------------------------------------------------------------------------
__global__ __launch_bounds__(256)
void mlp_wmma_kernel(const float* __restrict__ x16,
                     const float* __restrict__ W2,
                     const float* __restrict__ g2, const float* __restrict__ b2,
                     const float* __restrict__ m2, const float* __restrict__ v2,
                     const float* __restrict__ W3,
                     const float* __restrict__ g3, const float* __restrict__ b3,
                     const float* __restrict__ m3, const float* __restrict__ v3,
                     float* __restrict__ out) {
    __shared__ float sH[8][C4_ * 16];                // per-wave 32x16 f32 tile
    const int lane = threadIdx.x & 31;
    const int wave = threadIdx.x >> 5;
    const int tile = blockIdx.x * 8 + wave;          // 0 .. B*(M/16)-1
    const int b  = tile / (M_ / 16);
    const int m0 = (tile % (M_ / 16)) * 16;
    const int col = lane & 15;
    const int hi  = lane >> 4;                       // 0: K pair {0,1}, 1: {2,3}

    const float* __restrict__ xb = x16 + (size_t)b * C8_ * M_ + m0;

    // ---- GEMM1: h = W2 @ x16tile, K=16 in 4 WMMA steps, 2 o-tiles ----
    v8f acc0 = {0.f,0.f,0.f,0.f,0.f,0.f,0.f,0.f};
    v8f acc1 = {0.f,0.f,0.f,0.f,0.f,0.f,0.f,0.f};
    #pragma unroll
    for (int j = 0; j < 4; ++j) {
        const int kb = 4*j + 2*hi;
        v2f bm, a0, a1;
        bm.x = xb[(size_t)(kb + 0) * M_ + col];
        bm.y = xb[(size_t)(kb + 1) * M_ + col];
        a0.x = W2[(col)      * C8_ + kb + 0];
        a0.y = W2[(col)      * C8_ + kb + 1];
        a1.x = W2[(col + 16) * C8_ + kb + 0];
        a1.y = W2[(col + 16) * C8_ + kb + 1];
        acc0 = __builtin_amdgcn_wmma_f32_16x16x4_f32(false, a0, false, bm,
                                                     (short)0, acc0, false, false);
        acc1 = __builtin_amdgcn_wmma_f32_16x16x4_f32(false, a1, false, bm,
                                                     (short)0, acc1, false, false);
    }

    // ---- BN2 + GELU in D-layout, stage (c, col) tile to LDS ----
    float* __restrict__ hw = sH[wave];
    #pragma unroll
    for (int r = 0; r < 8; ++r) {
        const int o0 = r + hi * 8;        // channel in o-tile 0 (0..15)
        const int o1 = 16 + o0;           // channel in o-tile 1 (16..31)
        float s0 = g2[o0] * rsqrtf(v2[o0] + EPS_);
        float s1 = g2[o1] * rsqrtf(v2[o1] + EPS_);
        float y0 = acc0[r] * s0 + (b2[o0] - m2[o0] * s0);
        float y1 = acc1[r] * s1 + (b2[o1] - m2[o1] * s1);
        hw[o0 * 16 + col] = gelu_exact(y0);
        hw[o1 * 16 + col] = gelu_exact(y1);
    }
    __syncthreads();

    // ---- GEMM2: out = W3 @ h, K=32 in 8 WMMA steps, 8 o-tiles ----
    #pragma unroll
    for (int tt = 0; tt < 8; ++tt) {
        v8f acc = {0.f,0.f,0.f,0.f,0.f,0.f,0.f,0.f};
        #pragma unroll
        for (int j = 0; j < 8; ++j) {
            const int kb = 4*j + 2*hi;
            v2f bm, am;
            bm.x = hw[(kb + 0) * 16 + col];
            bm.y = hw[(kb + 1) * 16 + col];
            am.x = W3[(tt*16 + col) * C4_ + kb + 0];
            am.y = W3[(tt*16 + col) * C4_ + kb + 1];
            acc = __builtin_amdgcn_wmma_f32_16x16x4_f32(false, am, false, bm,
                                                        (short)0, acc, false, false);
        }
        #pragma unroll
        for (int r = 0; r < 8; ++r) {
            const int o = tt*16 + r + hi*8;
            float sc = g3[o] * rsqrtf(v3[o] + EPS_);
            float y  = acc[r] * sc + (b3[o] - m3[o] * sc);
            out[((size_t)b * C_ + o) * M_ + m0 + col] = y;
        }
    }
}

// ---------------------------------------------------------------------------
extern "C" void kernel_launch(void* const* d_in, const int* in_sizes, int n_in,
                              void* d_out, int out_size, void* d_ws, size_t ws_size,
                              hipStream_t stream) {
    const float* centroid = (const float*)d_in[0];
    const float* xyz      = (const float*)d_in[1];
    const float* radius   = (const float*)d_in[2];
    const float* dist     = (const float*)d_in[3];
    const float* W1       = (const float*)d_in[4];
    const float* b1       = (const float*)d_in[5];
    const float* bn1_g    = (const float*)d_in[6];
    const float* bn1_b    = (const float*)d_in[7];
    const float* bn1_m    = (const float*)d_in[8];
    const float* bn1_v    = (const float*)d_in[9];
    const float* W2       = (const float*)d_in[10];
    const float* bn2_g    = (const float*)d_in[11];
    const float* bn2_b    = (const float*)d_in[12];
    const float* bn2_m    = (const float*)d_in[13];
    const float* bn2_v    = (const float*)d_in[14];
    const float* W3       = (const float*)d_in[15];
    const float* bn3_g    = (const float*)d_in[16];
    const float* bn3_b    = (const float*)d_in[17];
    const float* bn3_m    = (const float*)d_in[18];
    const float* bn3_v    = (const float*)d_in[19];

    int*   idx = (int*)d_ws;
    float* x16 = (float*)((char*)d_ws + (size_t)B_ * M_ * K_ * sizeof(int));

    // 1 wave per dist row: B*M rows / 8 waves-per-block
    ball_query_kernel<<<(B_ * M_) / 8, 256, 0, stream>>>(dist, radius, idx);

    // 1 thread per (b, m)
    group_conv1_kernel<<<(B_ * M_) / 256, 256, 0, stream>>>(
        xyz, centroid, idx, W1, b1, bn1_g, bn1_b, bn1_m, bn1_v, x16);

    // 1 wave per 16-wide m-tile: B*(M/16) tiles / 8 waves-per-block
    mlp_wmma_kernel<<<(B_ * (M_ / 16)) / 8, 256, 0, stream>>>(
        x16, W2, bn2_g, bn2_b, bn2_m, bn2_v,
        W3, bn3_g, bn3_b, bn3_m, bn3_v, (float*)d_out);
}